// TENENCE_4269197492603
// MI455X (gfx1250) — compile-verified
//
#include <hip/hip_runtime.h>
#include <math.h>

// Problem constants (match reference)
#define NN 50000
#define EE 400000
#define TT 3
#define DD 128
#define INV_STD 0.9999950000374998f  // 1/sqrt(1 + 1e-5)

typedef __attribute__((ext_vector_type(2))) float v2f;
typedef __attribute__((ext_vector_type(8))) float v8f;

// ---------------------------------------------------------------------------
// GEMM: C[N x 128] = A[N x K] * W[K x 128], f32 WMMA 16x16x4.
// One wave computes a 16-row strip of C across all 128 columns (8 tiles).
// A-frag layout (16x4 f32): lane m = lane&15, half g = lane>>4,
//   vgpr0 = A[m][k+2g], vgpr1 = A[m][k+2g+1]
// B-frag layout (4x16 f32): lane n = lane&15, half g,
//   vgpr0 = W[k+2g][n], vgpr1 = W[k+2g+1][n]
// C/D layout (16x16 f32): lane n = lane&15, vgpr v -> row (8g + v)
// ---------------------------------------------------------------------------
template <int K>
__global__ __launch_bounds__(256) void gemm_wmma(const float* __restrict__ A,
                                                 const float* __restrict__ W,
                                                 float* __restrict__ C,
                                                 int nRows) {
  const int lane = threadIdx.x & 31;
  const int strip = blockIdx.x * 8 + (threadIdx.x >> 5);
  const int m0 = strip * 16;
  if (m0 >= nRows) return;  // wave-uniform: EXEC stays all-ones for WMMA
  const int g = lane >> 4;
  const int mr = lane & 15;
  const float* Arow = A + (size_t)(m0 + mr) * K;

  v8f acc[8] = {};
  for (int kk = 0; kk < K; kk += 4) {
    v2f a;
    a.x = Arow[kk + 2 * g + 0];
    a.y = Arow[kk + 2 * g + 1];
#pragma unroll
    for (int t = 0; t < 8; ++t) {
      const float* Wp = W + (size_t)(kk + 2 * g) * DD + t * 16 + mr;
      v2f b;
      b.x = Wp[0];
      b.y = Wp[DD];
      acc[t] = __builtin_amdgcn_wmma_f32_16x16x4_f32(
          false, a, false, b, (short)0, acc[t], false, false);
    }
  }
#pragma unroll
  for (int t = 0; t < 8; ++t) {
#pragma unroll
    for (int v = 0; v < 8; ++v) {
      C[(size_t)(m0 + 8 * g + v) * DD + t * 16 + mr] = acc[t][v];
    }
  }
}

// ---------------------------------------------------------------------------
// Elementwise / scatter kernels
// ---------------------------------------------------------------------------
__global__ void fill_f32(float* __restrict__ p, float v, int n) {
  int i = blockIdx.x * blockDim.x + threadIdx.x;
  if (i < n) p[i] = v;
}

__global__ void deg_count(const int* __restrict__ dst, float* __restrict__ indeg, int nE) {
  int e = blockIdx.x * blockDim.x + threadIdx.x;
  if (e < nE) atomicAdd(&indeg[dst[e]], 1.0f);
}

__global__ void deg_finalize(const float* __restrict__ indeg,
                             float* __restrict__ dinv1, float* __restrict__ f1v,
                             float* __restrict__ dinv2, float* __restrict__ f2v, int n) {
  int i = blockIdx.x * blockDim.x + threadIdx.x;
  if (i >= n) return;
  float d = indeg[i];
  float d1 = d + 1.0f;
  float d2 = d + 2.0f;
  dinv1[i] = rsqrtf(d1);
  f1v[i]   = 1.0f / d1;
  dinv2[i] = rsqrtf(d2);
  f2v[i]   = 2.0f / d2;
}

// agg[dst] += (dinv[src]*dinv[dst]) * xw[src];  32 threads per edge, float4 each
__global__ void gcn_scatter(const float* __restrict__ xw, const int* __restrict__ src,
                            const int* __restrict__ dst, const float* __restrict__ dinv,
                            float* __restrict__ agg, int nE) {
  int gid = blockIdx.x * blockDim.x + threadIdx.x;
  int e = gid >> 5;
  if (e >= nE) return;
  int cg = (gid & 31) << 2;
  int s = src[e], d = dst[e];
  float nrm = dinv[s] * dinv[d];
  const float4 v = *(const float4*)(xw + (size_t)s * DD + cg);
  float* ap = agg + (size_t)d * DD + cg;
  atomicAdd(ap + 0, nrm * v.x);
  atomicAdd(ap + 1, nrm * v.y);
  atomicAdd(ap + 2, nrm * v.z);
  atomicAdd(ap + 3, nrm * v.w);
}

// MPNN combine: v = BN(agg + (fill/deg)*xw + b); optional ReLU; dual-destination
__global__ void mpnn_combine(const float* __restrict__ agg, const float* __restrict__ xw,
                             const float* __restrict__ selfw, const float* __restrict__ bias,
                             const float* __restrict__ gamma, const float* __restrict__ beta,
                             float* __restrict__ out, int outStride,
                             float* __restrict__ out2, int doRelu, int n) {
  int i = blockIdx.x * blockDim.x + threadIdx.x;
  if (i >= n) return;
  int node = i >> 7, c = i & 127;
  float v = agg[i] + selfw[node] * xw[i] + bias[c];
  v = gamma[c] * (v * INV_STD) + beta[c];
  if (doRelu) v = fmaxf(v, 0.0f);
  out[(size_t)node * outStride + c] = v;
  if (out2) out2[i] = v;
}

// improved-GCN combine (fill=2): out = agg + (2/deg)*xw + b
__global__ void gcn2_combine(const float* __restrict__ agg, const float* __restrict__ xw,
                             const float* __restrict__ selfw, const float* __restrict__ bias,
                             float* __restrict__ out, int n) {
  int i = blockIdx.x * blockDim.x + threadIdx.x;
  if (i >= n) return;
  int node = i >> 7, c = i & 127;
  out[i] = agg[i] + selfw[node] * xw[i] + bias[c];
}

__global__ void set_last_seen(const int* __restrict__ src, float* __restrict__ ls,
                              float v, int nE) {
  int e = blockIdx.x * blockDim.x + threadIdx.x;
  if (e < nE) ls[src[e]] = v;  // duplicate writes store the same value
}

// zc[:,128+j] = cos(last_seen * 10^(-9 j / 127))
__global__ void time_encode(const float* __restrict__ ls, float* __restrict__ zc, int n) {
  int i = blockIdx.x * blockDim.x + threadIdx.x;
  if (i >= n) return;
  int node = i >> 7, j = i & 127;
  float w = expf(-2.302585092994046f * (9.0f * (float)j / 127.0f));
  zc[(size_t)node * (2 * DD) + DD + j] = cosf(ls[node] * w);
}

__global__ void gate_sigmoid(const float* __restrict__ a, const float* __restrict__ b,
                             float* __restrict__ out, int n) {
  int i = blockIdx.x * blockDim.x + threadIdx.x;
  if (i >= n) return;
  out[i] = 1.0f / (1.0f + expf(-(a[i] + b[i])));
}

__global__ void candidate(const float* __restrict__ a, const float* __restrict__ b,
                          const float* __restrict__ r, float* __restrict__ out, int n) {
  int i = blockIdx.x * blockDim.x + threadIdx.x;
  if (i >= n) return;
  out[i] = tanhf(a[i] + r[i] * b[i]);
}

__global__ void state_update(const float* __restrict__ c, const float* __restrict__ u,
                             float* __restrict__ st, float* __restrict__ so, int n) {
  int i = blockIdx.x * blockDim.x + threadIdx.x;
  if (i >= n) return;
  float s = (1.0f - u[i]) * c[i] + u[i] * st[i];
  st[i] = s;
  so[i] = s;
}

__global__ void bias_add(const float* __restrict__ xw, const float* __restrict__ bias,
                         float* __restrict__ out, int n) {
  int i = blockIdx.x * blockDim.x + threadIdx.x;
  if (i >= n) return;
  out[i] = xw[i] + bias[i & 127];
}

// ---------------------------------------------------------------------------
// Host orchestration
// ---------------------------------------------------------------------------
static inline void run_gemm128(const float* A, const float* W, float* C, hipStream_t s) {
  const int strips = NN / 16;  // 3125
  gemm_wmma<128><<<dim3((strips + 7) / 8), dim3(256), 0, s>>>(A, W, C, NN);
}
static inline void run_gemm256(const float* A, const float* W, float* C, hipStream_t s) {
  const int strips = NN / 16;
  gemm_wmma<256><<<dim3((strips + 7) / 8), dim3(256), 0, s>>>(A, W, C, NN);
}

extern "C" void kernel_launch(void* const* d_in, const int* in_sizes, int n_in,
                              void* d_out, int out_size, void* d_ws, size_t ws_size,
                              hipStream_t stream) {
  const float* x      = (const float*)d_in[0];
  const int*   esrc   = (const int*)d_in[1];
  const int*   edst   = (const int*)d_in[2];
  const float* mpnnW  = (const float*)d_in[3];
  const float* mpnnB  = (const float*)d_in[4];
  const float* bnG    = (const float*)d_in[5];
  const float* bnB    = (const float*)d_in[6];
  const float* Wi     = (const float*)d_in[7];
  const float* bi     = (const float*)d_in[8];
  const float* Ws     = (const float*)d_in[9];
  const float* bs     = (const float*)d_in[10];
  const float* decW   = (const float*)d_in[11];
  const float* decB   = (const float*)d_in[12];
  const float* predW  = (const float*)d_in[13];
  const float* predB  = (const float*)d_in[14];
  float* out = (float*)d_out;

  // workspace carve-up (all offsets are multiples of NN floats -> 16B aligned)
  float* ws = (float*)d_ws;
  size_t off = 0;
  auto carve = [&](size_t nf) { float* p = ws + off; off += nf; return p; };
  float* indeg    = carve(NN);
  float* dinv1    = carve(NN);
  float* f1v      = carve(NN);
  float* dinv2    = carve(NN);
  float* f2v      = carve(NN);
  float* lastseen = carve(NN);
  float* xw    = carve((size_t)NN * DD);
  float* agg   = carve((size_t)NN * DD);
  float* zc    = carve((size_t)NN * 2 * DD);
  float* gA    = carve((size_t)NN * DD);
  float* gB    = carve((size_t)NN * DD);
  float* rbuf  = carve((size_t)NN * DD);
  float* ubuf  = carve((size_t)NN * DD);
  float* state = carve((size_t)NN * DD);
  (void)in_sizes; (void)n_in; (void)out_size; (void)ws_size;

  const int NC = NN * DD;  // 6.4M
  dim3 b256(256);
  auto g1 = [](int n) { return dim3((n + 255) / 256); };

  fill_f32<<<g1(NC), b256, 0, stream>>>(state, 0.0f, NC);
  fill_f32<<<g1(NN), b256, 0, stream>>>(lastseen, 0.0f, NN);

  for (int k = 0; k < TT; ++k) {
    const int* sk = esrc + (size_t)k * EE;
    const int* dk = edst + (size_t)k * EE;

    // degrees and normalization factors
    fill_f32<<<g1(NN), b256, 0, stream>>>(indeg, 0.0f, NN);
    deg_count<<<g1(EE), b256, 0, stream>>>(dk, indeg, EE);
    deg_finalize<<<g1(NN), b256, 0, stream>>>(indeg, dinv1, f1v, dinv2, f2v, NN);

    // ---- MPNN encoder: 3 GCN layers (fill=1) + BN (+ ReLU on first two) ----
    const float* hin = x + (size_t)k * NN * DD;
    float* zping[2] = {gA, gB};
    for (int l = 0; l < 3; ++l) {
      const float* Ain = (l == 0) ? hin : zping[(l - 1) & 1];
      run_gemm128(Ain, mpnnW + (size_t)l * DD * DD, xw, stream);
      fill_f32<<<g1(NC), b256, 0, stream>>>(agg, 0.0f, NC);
      gcn_scatter<<<g1(EE * 32), b256, 0, stream>>>(xw, sk, dk, dinv1, agg, EE);
      if (l < 2) {
        mpnn_combine<<<g1(NC), b256, 0, stream>>>(
            agg, xw, f1v, mpnnB + l * DD, bnG + l * DD, bnB + l * DD,
            zping[l & 1], DD, (float*)nullptr, 1, NC);
      } else {
        // write z into zc[:, :128] (stride 256) and to Z_enc output
        mpnn_combine<<<g1(NC), b256, 0, stream>>>(
            agg, xw, f1v, mpnnB + l * DD, bnG + l * DD, bnB + l * DD,
            zc, 2 * DD, out + (size_t)(TT + k) * NN * DD, 0, NC);
      }
    }

    // ---- last_seen scatter-set + GraphMixer time encoding into zc[:,128:] ----
    set_last_seen<<<g1(EE), b256, 0, stream>>>(sk, lastseen, (float)(k + 1), EE);
    time_encode<<<g1(NC), b256, 0, stream>>>(lastseen, zc, NC);

    // ---- GGRU: 3 gates, each gate = gcn2(zc,Wi) + gcn2(state,Ws) ----
    for (int gi = 0; gi < 3; ++gi) {
      // input branch (K = 256, fill = 2)
      run_gemm256(zc, Wi + (size_t)gi * 2 * DD * DD, xw, stream);
      fill_f32<<<g1(NC), b256, 0, stream>>>(agg, 0.0f, NC);
      gcn_scatter<<<g1(EE * 32), b256, 0, stream>>>(xw, sk, dk, dinv2, agg, EE);
      gcn2_combine<<<g1(NC), b256, 0, stream>>>(agg, xw, f2v, bi + gi * DD, gA, NC);
      // state branch (K = 128, fill = 2)
      run_gemm128(state, Ws + (size_t)gi * DD * DD, xw, stream);
      fill_f32<<<g1(NC), b256, 0, stream>>>(agg, 0.0f, NC);
      gcn_scatter<<<g1(EE * 32), b256, 0, stream>>>(xw, sk, dk, dinv2, agg, EE);
      gcn2_combine<<<g1(NC), b256, 0, stream>>>(agg, xw, f2v, bs + gi * DD, gB, NC);

      if (gi == 0)      gate_sigmoid<<<g1(NC), b256, 0, stream>>>(gA, gB, rbuf, NC);
      else if (gi == 1) gate_sigmoid<<<g1(NC), b256, 0, stream>>>(gA, gB, ubuf, NC);
      else              candidate<<<g1(NC), b256, 0, stream>>>(gA, gB, rbuf, gA, NC);
    }
    // state = (1-u)*c + u*state ; also emit states output
    state_update<<<g1(NC), b256, 0, stream>>>(gA, ubuf, state,
                                              out + (size_t)k * NN * DD, NC);

    // ---- decoder + predictor heads ----
    run_gemm128(state, decW, xw, stream);
    bias_add<<<g1(NC), b256, 0, stream>>>(xw, decB, out + (size_t)(2 * TT + k) * NN * DD, NC);
    run_gemm128(state, predW, xw, stream);
    bias_add<<<g1(NC), b256, 0, stream>>>(xw, predB, out + (size_t)(3 * TT + k) * NN * DD, NC);
  }
}